// MessagePasser_44367012168461
// MI455X (gfx1250) — compile-verified
//
#include <hip/hip_runtime.h>

typedef __attribute__((ext_vector_type(2))) float v2f;
typedef __attribute__((ext_vector_type(8))) float v8f;

#define Bn    8
#define Nn    64
#define En    1024
#define INn   32
#define MSGn  64
#define BONDn 16
#define CINn  80     // 2*IN + BOND
#define HIDn  64
#define OUTn  32
#define VINn  96     // IN + MSG
#define NEG_SLOPE 0.01f

__device__ __forceinline__ float leaky(float x) { return x >= 0.0f ? x : NEG_SLOPE * x; }

// ---------------------------------------------------------------------------
// init: Weff = Wl + Wg/64 (the one-hot algebra collapses to this), msgs = 0,
// and pass bonds straight through to the second output.
// ---------------------------------------------------------------------------
__global__ __launch_bounds__(256) void init_kernel(
    const float* __restrict__ Wl, const float* __restrict__ Wg,
    const float* __restrict__ bonds,
    float* __restrict__ weff, float* __restrict__ msgs,
    float* __restrict__ out_bonds)
{
  int tid = blockIdx.x * blockDim.x + threadIdx.x;
  if (tid < MSGn * CINn)      weff[tid] = Wl[tid] + (1.0f / 64.0f) * Wg[tid];
  if (tid < Bn * Nn * MSGn)   msgs[tid] = 0.0f;
  if (tid < Bn * En * BONDn)  out_bonds[tid] = bonds[tid];
}

// ---------------------------------------------------------------------------
// Edge kernel: one wave computes a 16-edge x 64-msg tile via WMMA f32 16x16x4,
// fuses bias + leaky + attention (sigmoid of lat.wa) + atomic scatter-add.
// Gathered GEMM: M = B*E = 8192 rows, K = 80, N = 64.
// ---------------------------------------------------------------------------
__global__ __launch_bounds__(128) void edge_kernel(
    const float* __restrict__ sites, const float* __restrict__ bonds,
    const int* __restrict__ idx1, const int* __restrict__ idx2,
    const float* __restrict__ weff, const float* __restrict__ b_eq,
    const float* __restrict__ wa, const float* __restrict__ ba,
    float* __restrict__ msgs)
{
  const int lane = threadIdx.x & 31;
  const int wave = threadIdx.x >> 5;
  const int tile = blockIdx.x * 4 + wave;   // 0..511, 16 rows of (b,e) each
  const int half = lane >> 4;               // 0 or 1
  const int mr   = lane & 15;

  // A-side row handled by this lane (A fragment: lane 16h+m holds row m)
  const int m  = tile * 16 + mr;            // global row in [0, B*E)
  const int b  = m >> 10;                   // / En
  const int e  = m & (En - 1);
  const int i1 = idx1[e];
  const int i2 = idx2[e];
  const float* s1 = sites + (size_t)(b * Nn + i1) * INn;
  const float* s2 = sites + (size_t)(b * Nn + i2) * INn;
  const float* bd = bonds + (size_t)(b * En + e) * BONDn;

  v8f acc[4] = {};                          // 4 column tiles of 16 -> N=64

#pragma unroll
  for (int k0 = 0; k0 < CINn; k0 += 4) {
    const int c = k0 + 2 * half;            // this lane's K pair
    v2f a;
    if (c < INn)              a = (v2f){ s1[c],            s1[c + 1] };
    else if (c < 2 * INn)     a = (v2f){ s2[c - INn],      s2[c - INn + 1] };
    else                      a = (v2f){ bd[c - 2 * INn],  bd[c - 2 * INn + 1] };
#pragma unroll
    for (int t = 0; t < 4; ++t) {
      const int n   = t * 16 + mr;          // B fragment: row K=c, col n
      const int off = n * CINn + c;         // B[k][n] = Weff[n][k]
      v2f bf = (v2f){ weff[off], weff[off + 1] };
      acc[t] = __builtin_amdgcn_wmma_f32_16x16x4_f32(
          false, a, false, bf, (short)0, acc[t], false, false);
    }
  }

  // bias + leaky. C layout: element j -> row (j + 8*half), col (t*16 + mr)
  float wav[4];
#pragma unroll
  for (int t = 0; t < 4; ++t) {
    const int n = t * 16 + mr;
    const float bb = b_eq[n];
    wav[t] = wa[n];
#pragma unroll
    for (int j = 0; j < 8; ++j) acc[t][j] = leaky(acc[t][j] + bb);
  }

  // attention: per-row dot(lat, wa) -> reduce across the 16 lanes of the half
  float p[8];
#pragma unroll
  for (int j = 0; j < 8; ++j)
    p[j] = acc[0][j] * wav[0] + acc[1][j] * wav[1] +
           acc[2][j] * wav[2] + acc[3][j] * wav[3];
#pragma unroll
  for (int off = 8; off >= 1; off >>= 1) {
#pragma unroll
    for (int j = 0; j < 8; ++j) p[j] += __shfl_xor(p[j], off, 32);
  }
  const float ba0 = ba[0];
  float att[8];
#pragma unroll
  for (int j = 0; j < 8; ++j) att[j] = 1.0f / (1.0f + __expf(-(p[j] + ba0)));

  // scatter: msgs[b, idx2[e], n] += att * lat
  const int ebase = (tile & 63) * 16;       // e of row 0 of this tile
  const int brow  = tile >> 6;
#pragma unroll
  for (int j = 0; j < 8; ++j) {
    const int r  = j + 8 * half;            // row within tile for C element j
    const int er = ebase + r;
    float* base = msgs + (size_t)(brow * Nn + idx2[er]) * MSGn;
    const float aj = att[j];
#pragma unroll
    for (int t = 0; t < 4; ++t)
      atomicAdd(base + t * 16 + mr, aj * acc[t][j]);
  }
}

// ---------------------------------------------------------------------------
// Node kernel: h = leaky([sites|msgs] @ W1^T + b1); out = sites + leaky(h@W2^T + b2)
// Two chained WMMA GEMMs; h re-staged through LDS (C/D layout -> A layout).
// M = B*N = 512 rows.
// ---------------------------------------------------------------------------
__global__ __launch_bounds__(128) void node_kernel(
    const float* __restrict__ sites, const float* __restrict__ msgs,
    const float* __restrict__ W1, const float* __restrict__ b1,
    const float* __restrict__ W2, const float* __restrict__ b2,
    float* __restrict__ out)
{
  __shared__ float hbuf[4][16 * HIDn];      // one 16x64 tile per wave
  const int lane = threadIdx.x & 31;
  const int wave = threadIdx.x >> 5;
  const int tile = blockIdx.x * 4 + wave;   // 0..31
  const int half = lane >> 4;
  const int mr   = lane & 15;

  const int row = tile * 16 + mr;           // A row (flat b*N + n)
  const float* sr = sites + (size_t)row * INn;
  const float* mg = msgs  + (size_t)row * MSGn;

  // GEMM1: K = 96, N = 64
  v8f acc[4] = {};
#pragma unroll
  for (int k0 = 0; k0 < VINn; k0 += 4) {
    const int c = k0 + 2 * half;
    v2f a = (c < INn) ? (v2f){ sr[c], sr[c + 1] }
                      : (v2f){ mg[c - INn], mg[c - INn + 1] };
#pragma unroll
    for (int t = 0; t < 4; ++t) {
      const int off = (t * 16 + mr) * VINn + c;   // B[k][n] = W1[n][k]
      v2f bf = (v2f){ W1[off], W1[off + 1] };
      acc[t] = __builtin_amdgcn_wmma_f32_16x16x4_f32(
          false, a, false, bf, (short)0, acc[t], false, false);
    }
  }

  // bias + leaky, stage h into LDS in row-major for the next A fragment
#pragma unroll
  for (int t = 0; t < 4; ++t) {
    const float bb = b1[t * 16 + mr];
#pragma unroll
    for (int j = 0; j < 8; ++j)
      hbuf[wave][(j + 8 * half) * HIDn + t * 16 + mr] = leaky(acc[t][j] + bb);
  }
  __syncthreads();

  // GEMM2: K = 64, N = 32
  v8f o[2] = {};
#pragma unroll
  for (int k0 = 0; k0 < HIDn; k0 += 4) {
    const int c = k0 + 2 * half;
    v2f a = (v2f){ hbuf[wave][mr * HIDn + c], hbuf[wave][mr * HIDn + c + 1] };
#pragma unroll
    for (int t = 0; t < 2; ++t) {
      const int off = (t * 16 + mr) * HIDn + c;   // B[k][n] = W2[n][k]
      v2f bf = (v2f){ W2[off], W2[off + 1] };
      o[t] = __builtin_amdgcn_wmma_f32_16x16x4_f32(
          false, a, false, bf, (short)0, o[t], false, false);
    }
  }

  // residual add + store (OUT == IN == 32)
#pragma unroll
  for (int t = 0; t < 2; ++t) {
    const int ncol = t * 16 + mr;
    const float bb = b2[ncol];
#pragma unroll
    for (int j = 0; j < 8; ++j) {
      const int grow = tile * 16 + j + 8 * half;
      out[(size_t)grow * OUTn + ncol] =
          sites[(size_t)grow * INn + ncol] + leaky(o[t][j] + bb);
    }
  }
}

// ---------------------------------------------------------------------------
extern "C" void kernel_launch(void* const* d_in, const int* in_sizes, int n_in,
                              void* d_out, int out_size, void* d_ws, size_t ws_size,
                              hipStream_t stream) {
  const float* sites = (const float*)d_in[0];
  const float* bonds = (const float*)d_in[1];
  const int*   idx1  = (const int*)  d_in[2];
  const int*   idx2  = (const int*)  d_in[3];
  const float* Wl    = (const float*)d_in[4];
  const float* Wg    = (const float*)d_in[5];
  const float* b_eq  = (const float*)d_in[6];
  const float* wa    = (const float*)d_in[7];
  const float* ba    = (const float*)d_in[8];
  const float* W1    = (const float*)d_in[9];
  const float* b1    = (const float*)d_in[10];
  const float* W2    = (const float*)d_in[11];
  const float* b2    = (const float*)d_in[12];

  float* out       = (float*)d_out;                 // sites_out: 8*64*32
  float* out_bonds = out + Bn * Nn * INn;           // bonds pass-through
  float* msgs      = (float*)d_ws;                  // 8*64*64 floats
  float* weff      = msgs + Bn * Nn * MSGn;         // 64*80 floats

  // 1) Weff precompute + msgs zero + bonds copy (131072 elems max)
  init_kernel<<<512, 256, 0, stream>>>(Wl, Wg, bonds, weff, msgs, out_bonds);
  // 2) gathered edge GEMM + attention + scatter: 512 tiles, 4 waves/block
  edge_kernel<<<128, 128, 0, stream>>>(sites, bonds, idx1, idx2, weff,
                                       b_eq, wa, ba, msgs);
  // 3) node MLP: 32 tiles, 4 waves/block
  node_kernel<<<8, 128, 0, stream>>>(sites, msgs, W1, b1, W2, b2, out);
}